// MambaBlock_35364760715772
// MI455X (gfx1250) — compile-verified
//
#include <hip/hip_runtime.h>
#include <hip/hip_bf16.h>
#include <math.h>

// ---------------------------------------------------------------------------
// Mamba-v3 block for MI455X (gfx1250, wave32).
// GEMMs via V_WMMA_F32_16X16X4_F32 (fully unrolled K, no exec divergence in
// the hot loop); selective scan via chunked parallel scan (64 chunks x 192 ch).
// ---------------------------------------------------------------------------

typedef __attribute__((ext_vector_type(2))) float v2f;
typedef __attribute__((ext_vector_type(8))) float v8f;

static constexpr int L      = 4096;
static constexpr int DIM    = 96;
static constexpr int DIN    = 192;   // DIM * EXPAND
static constexpr int DPROJ  = 384;   // 2 * DIN
static constexpr int DSTATE = 16;
static constexpr int DTRANK = 6;
static constexpr int NPAD   = 48;    // x-proj N (38) padded to 48
static constexpr int XPN    = 38;    // DTRANK + 2*DSTATE
static constexpr int NCH    = 64;    // scan chunks
static constexpr int CHL    = 64;    // chunk length (L / NCH)

__device__ __forceinline__ float sigmoidf_(float x) {
    return 1.0f / (1.0f + __expf(-x));
}
__device__ __forceinline__ float siluf_(float x) {
    return x * sigmoidf_(x);
}
__device__ __forceinline__ float softplusf_(float x) {
    return (x > 20.0f) ? x : log1pf(__expf(x));
}
// Branch time mapping (scan position p -> original sequence index l).
// mode 0: identity (forward); 1: reversed; 2: 64x64 time transpose (slice).
// All three are involutions.
__device__ __forceinline__ int map_pos(int p, int mode) {
    if (mode == 0) return p;
    if (mode == 1) return (L - 1) - p;
    return ((p & 63) << 6) | (p >> 6);
}

// ---------------------------------------------------------------------------
// Generic fp32 WMMA GEMM:  C[M x N] = act( A[M x K] * W[N x K]^T (+ res) )
// One wave (32 lanes) per 16x16 output tile. grid = (M/16, Npad/16).
// K is a compile-time constant -> fully unrolled WMMA chain, loads batched.
// Padded columns (ncol >= realN) read a clamped-but-valid weight row; their
// accumulator columns are garbage but never stored, so no masking/divergence
// is needed in the hot loop.
// ACT: 0 = none, 1 = swish (x*sigmoid(x)), 2 = add residual.
// ---------------------------------------------------------------------------
template <int ACT, int K>
__global__ void __launch_bounds__(32)
gemm_f32_wmma(const float* __restrict__ A, int lda,
              const float* __restrict__ W, int ldw,
              float* __restrict__ C, int ldc,
              int realN,
              const float* __restrict__ res, int ldres)
{
    const int lane  = threadIdx.x;       // 0..31
    const int rc    = lane & 15;         // A row / B col within tile
    const int khalf = (lane >> 4) * 2;   // 0 or 2
    const int arow  = blockIdx.x * 16 + rc;
    const int ncol  = blockIdx.y * 16 + rc;
    const int nclmp = (ncol < realN) ? ncol : (realN - 1);

    const float* aptr = A + (size_t)arow  * lda + khalf;
    const float* wptr = W + (size_t)nclmp * ldw + khalf;

    v8f acc = {};
#pragma unroll
    for (int k0 = 0; k0 < K; k0 += 4) {
        v2f a, b;
        a.x = aptr[k0];
        a.y = aptr[k0 + 1];
        b.x = wptr[k0];
        b.y = wptr[k0 + 1];
        acc = __builtin_amdgcn_wmma_f32_16x16x4_f32(
            false, a, false, b, (short)0, acc, false, false);
    }

    if (ncol < realN) {
        const int mbase = blockIdx.x * 16 + (lane >> 4) * 8;
#pragma unroll
        for (int r = 0; r < 8; ++r) {
            float v = acc[r];
            const int row = mbase + r;
            if (ACT == 1) v = v * sigmoidf_(v);
            if (ACT == 2) v += res[(size_t)row * ldres + ncol];
            C[(size_t)row * ldc + ncol] = v;
        }
    }
}

// ---------------------------------------------------------------------------
// LayerNorm over 96 features; one wave per row (3 features/lane).
// DOUBLE=true applies a second LN (mnorm) on the normalized result.
// ---------------------------------------------------------------------------
__device__ __forceinline__ float wave_sum32(float v) {
#pragma unroll
    for (int off = 16; off >= 1; off >>= 1)
        v += __shfl_xor(v, off, 32);
    return v;
}

template <bool DOUBLE>
__global__ void __launch_bounds__(128)
layernorm_kernel(const float* __restrict__ X,
                 const float* __restrict__ w1, const float* __restrict__ b1,
                 const float* __restrict__ w2, const float* __restrict__ b2,
                 float* __restrict__ Y)
{
    const int row  = blockIdx.x * 4 + (threadIdx.x >> 5);
    const int lane = threadIdx.x & 31;
    if (row >= L) return;
    const float* xr = X + (size_t)row * DIM;

    float v[3];
#pragma unroll
    for (int i = 0; i < 3; ++i) v[i] = xr[lane + 32 * i];

    float m = wave_sum32(v[0] + v[1] + v[2]) * (1.0f / DIM);
    float q = 0.0f;
#pragma unroll
    for (int i = 0; i < 3; ++i) { float d = v[i] - m; q += d * d; }
    float rs = rsqrtf(wave_sum32(q) * (1.0f / DIM) + 1e-5f);
#pragma unroll
    for (int i = 0; i < 3; ++i)
        v[i] = (v[i] - m) * rs * w1[lane + 32 * i] + b1[lane + 32 * i];

    if (DOUBLE) {
        m = wave_sum32(v[0] + v[1] + v[2]) * (1.0f / DIM);
        q = 0.0f;
#pragma unroll
        for (int i = 0; i < 3; ++i) { float d = v[i] - m; q += d * d; }
        rs = rsqrtf(wave_sum32(q) * (1.0f / DIM) + 1e-5f);
#pragma unroll
        for (int i = 0; i < 3; ++i)
            v[i] = (v[i] - m) * rs * w2[lane + 32 * i] + b2[lane + 32 * i];
    }

    float* yr = Y + (size_t)row * DIM;
#pragma unroll
    for (int i = 0; i < 3; ++i) yr[lane + 32 * i] = v[i];
}

// ---------------------------------------------------------------------------
// Causal depthwise conv (taps=4) + SiLU, reading xz (L x 384, x part = cols
// 0..191) through the branch time-mapping. Output xc in scan order (L x 192).
// grid = L blocks, 192 threads (one channel each).
// ---------------------------------------------------------------------------
__global__ void __launch_bounds__(192)
conv_silu_kernel(const float* __restrict__ xz,
                 const float* __restrict__ cw, const float* __restrict__ cb,
                 float* __restrict__ xc, int mode)
{
    const int p = blockIdx.x;
    const int c = threadIdx.x;
    float acc = cb[c];
#pragma unroll
    for (int k = 0; k < 4; ++k) {
        const int q = p - 3 + k;
        if (q >= 0)
            acc += cw[c * 4 + k] * xz[(size_t)map_pos(q, mode) * DPROJ + c];
    }
    xc[(size_t)p * DIN + c] = siluf_(acc);
}

// ---------------------------------------------------------------------------
// delta[p,c] = softplus( dbl[p,0:6] . dtw[c,:] + dtb[c] )
// ---------------------------------------------------------------------------
__global__ void __launch_bounds__(192)
delta_kernel(const float* __restrict__ dbl,
             const float* __restrict__ dtw, const float* __restrict__ dtb,
             float* __restrict__ delta)
{
    const int p = blockIdx.x;
    const int c = threadIdx.x;
    float t = dtb[c];
#pragma unroll
    for (int r = 0; r < DTRANK; ++r)
        t = fmaf(dbl[(size_t)p * NPAD + r], dtw[c * DTRANK + r], t);
    delta[(size_t)p * DIN + c] = softplusf_(t);
}

// ---------------------------------------------------------------------------
// Scan phase 1: per (chunk, channel) compute per-state (prod dA, h_end|h0=0).
// grid = 64 chunks, 192 threads (one channel each).
// ---------------------------------------------------------------------------
__global__ void __launch_bounds__(192)
scan_phase1(const float* __restrict__ delta, const float* __restrict__ xc,
            const float* __restrict__ dbl,  const float* __restrict__ Alog,
            float* __restrict__ chunkA, float* __restrict__ chunkH)
{
    const int ch = blockIdx.x;
    const int c  = threadIdx.x;
    float Ac[DSTATE], h[DSTATE], ap[DSTATE];
#pragma unroll
    for (int n = 0; n < DSTATE; ++n) {
        Ac[n] = -__expf(Alog[c * DSTATE + n]);
        h[n] = 0.0f; ap[n] = 1.0f;
    }
    for (int t = 0; t < CHL; ++t) {
        const int p = ch * CHL + t;
        const float dl  = delta[(size_t)p * DIN + c];
        const float u   = xc[(size_t)p * DIN + c];
        const float dbu = dl * u;
        const float* bm = dbl + (size_t)p * NPAD + DTRANK;
#pragma unroll
        for (int n = 0; n < DSTATE; ++n) {
            const float a = __expf(dl * Ac[n]);
            ap[n] *= a;
            h[n] = fmaf(a, h[n], dbu * bm[n]);
        }
    }
    float* pa = chunkA + ((size_t)ch * DIN + c) * DSTATE;
    float* ph = chunkH + ((size_t)ch * DIN + c) * DSTATE;
#pragma unroll
    for (int n = 0; n < DSTATE; ++n) { pa[n] = ap[n]; ph[n] = h[n]; }
}

// ---------------------------------------------------------------------------
// Carry scan across the 64 chunks, one thread per (channel, state) = 3072.
// carryIn[k] = state entering chunk k (exclusive prefix).
// ---------------------------------------------------------------------------
__global__ void __launch_bounds__(256)
scan_carry(const float* __restrict__ chunkA, const float* __restrict__ chunkH,
           float* __restrict__ carryIn)
{
    const int idx = blockIdx.x * blockDim.x + threadIdx.x;
    if (idx >= DIN * DSTATE) return;
    const int c = idx >> 4;
    const int n = idx & 15;
    float carry = 0.0f;
    for (int k = 0; k < NCH; ++k) {
        const size_t off = ((size_t)k * DIN + c) * DSTATE + n;
        carryIn[off] = carry;
        carry = fmaf(chunkA[off], carry, chunkH[off]);
    }
}

// ---------------------------------------------------------------------------
// Scan phase 2: re-run chunk with carried state; fuse y = h.C + u*D, gate by
// silu(z), and scatter through the inverse time-mapping into y_total (L x192).
// ---------------------------------------------------------------------------
template <bool FIRST>
__global__ void __launch_bounds__(192)
scan_phase2(const float* __restrict__ delta, const float* __restrict__ xc,
            const float* __restrict__ dbl,  const float* __restrict__ Alog,
            const float* __restrict__ Dp,   const float* __restrict__ xz,
            const float* __restrict__ carryIn,
            float* __restrict__ y_total, int mode)
{
    const int ch = blockIdx.x;
    const int c  = threadIdx.x;
    float Ac[DSTATE], h[DSTATE];
    const float* ci = carryIn + ((size_t)ch * DIN + c) * DSTATE;
#pragma unroll
    for (int n = 0; n < DSTATE; ++n) {
        Ac[n] = -__expf(Alog[c * DSTATE + n]);
        h[n] = ci[n];
    }
    const float Dc = Dp[c];
    for (int t = 0; t < CHL; ++t) {
        const int p = ch * CHL + t;
        const int l = map_pos(p, mode);
        const float dl  = delta[(size_t)p * DIN + c];
        const float u   = xc[(size_t)p * DIN + c];
        const float dbu = dl * u;
        const float* bm = dbl + (size_t)p * NPAD + DTRANK;
        const float* cm = bm + DSTATE;
        float y = 0.0f;
#pragma unroll
        for (int n = 0; n < DSTATE; ++n) {
            const float a = __expf(dl * Ac[n]);
            h[n] = fmaf(a, h[n], dbu * bm[n]);
            y = fmaf(h[n], cm[n], y);
        }
        y = fmaf(u, Dc, y);
        const float z = xz[(size_t)l * DPROJ + DIN + c];
        y *= siluf_(z);
        float* dst = y_total + (size_t)l * DIN + c;
        if (FIRST) *dst = y;
        else       *dst += y;
    }
}

// ---------------------------------------------------------------------------
// Host-side launch sequence.
// ---------------------------------------------------------------------------
extern "C" void kernel_launch(void* const* d_in, const int* in_sizes, int n_in,
                              void* d_out, int out_size, void* d_ws, size_t ws_size,
                              hipStream_t stream)
{
    (void)in_sizes; (void)n_in; (void)out_size; (void)ws_size;

    const float* x       = (const float*)d_in[0];
    const float* norm1_w = (const float*)d_in[4];
    const float* norm2_w = (const float*)d_in[5];
    const float* mnorm_w = (const float*)d_in[6];
    const float* norm1_b = (const float*)d_in[7];
    const float* norm2_b = (const float*)d_in[8];
    const float* mnorm_b = (const float*)d_in[9];
    const float* in_w    = (const float*)d_in[10];
    const float* out_w   = (const float*)d_in[11];
    const float* mlp_w0  = (const float*)d_in[33];
    const float* mlp_w1  = (const float*)d_in[34];

    float* ws = (float*)d_ws;
    float* xm      = ws;                  ws += (size_t)L * DIM;
    float* xz      = ws;                  ws += (size_t)L * DPROJ;
    float* xc      = ws;                  ws += (size_t)L * DIN;
    float* dbl     = ws;                  ws += (size_t)L * NPAD;
    float* delta   = ws;                  ws += (size_t)L * DIN;
    float* chunkA  = ws;                  ws += (size_t)NCH * DIN * DSTATE;
    float* chunkH  = ws;                  ws += (size_t)NCH * DIN * DSTATE;
    float* carryIn = ws;                  ws += (size_t)NCH * DIN * DSTATE;
    float* y_total = ws;                  ws += (size_t)L * DIN;
    float* x2      = ws;                  ws += (size_t)L * DIM;
    float* hln     = ws;                  ws += (size_t)L * DIM;
    float* h1      = ws;                  ws += (size_t)L * DPROJ;

    // 1) fused double layernorm: xm = LN(LN(x, norm1), mnorm)
    layernorm_kernel<true><<<L / 4, 128, 0, stream>>>(
        x, norm1_w, norm1_b, mnorm_w, mnorm_b, xm);

    // 2) in-projection: xz = xm @ in_w^T   (4096x96 * 96->384)
    gemm_f32_wmma<0, DIM><<<dim3(L / 16, DPROJ / 16), 32, 0, stream>>>(
        xm, DIM, in_w, DIM, xz, DPROJ, DPROJ, nullptr, 0);

    // 3) three scan branches (forward / reversed / slice-transposed)
    for (int s = 0; s < 3; ++s) {
        const float* cw   = (const float*)d_in[12 + 7 * s + 0];
        const float* cb   = (const float*)d_in[12 + 7 * s + 1];
        const float* xpw  = (const float*)d_in[12 + 7 * s + 2];
        const float* dtw  = (const float*)d_in[12 + 7 * s + 3];
        const float* dtb  = (const float*)d_in[12 + 7 * s + 4];
        const float* Alog = (const float*)d_in[12 + 7 * s + 5];
        const float* Dp   = (const float*)d_in[12 + 7 * s + 6];

        conv_silu_kernel<<<L, DIN, 0, stream>>>(xz, cw, cb, xc, s);

        // x-projection: dbl = xc @ xpw^T   (4096x192 * 192->38, padded 48)
        gemm_f32_wmma<0, DIN><<<dim3(L / 16, NPAD / 16), 32, 0, stream>>>(
            xc, DIN, xpw, DIN, dbl, NPAD, XPN, nullptr, 0);

        delta_kernel<<<L, DIN, 0, stream>>>(dbl, dtw, dtb, delta);

        scan_phase1<<<NCH, DIN, 0, stream>>>(delta, xc, dbl, Alog, chunkA, chunkH);
        scan_carry<<<(DIN * DSTATE + 255) / 256, 256, 0, stream>>>(chunkA, chunkH, carryIn);
        if (s == 0)
            scan_phase2<true><<<NCH, DIN, 0, stream>>>(
                delta, xc, dbl, Alog, Dp, xz, carryIn, y_total, s);
        else
            scan_phase2<false><<<NCH, DIN, 0, stream>>>(
                delta, xc, dbl, Alog, Dp, xz, carryIn, y_total, s);
    }

    // 4) out-projection + residual: x2 = x + y_total @ out_w^T
    gemm_f32_wmma<2, DIN><<<dim3(L / 16, DIM / 16), 32, 0, stream>>>(
        y_total, DIN, out_w, DIN, x2, DIM, DIM, x, DIM);

    // 5) MLP: hln = LN(x2); h1 = swish(hln @ w0^T); out = x2 + h1 @ w1^T
    layernorm_kernel<false><<<L / 4, 128, 0, stream>>>(
        x2, norm2_w, norm2_b, nullptr, nullptr, hln);

    gemm_f32_wmma<1, DIM><<<dim3(L / 16, DPROJ / 16), 32, 0, stream>>>(
        hln, DIM, mlp_w0, DIM, h1, DPROJ, DPROJ, nullptr, 0);

    gemm_f32_wmma<2, DPROJ><<<dim3(L / 16, DIM / 16), 32, 0, stream>>>(
        h1, DPROJ, mlp_w1, DPROJ, (float*)d_out, DIM, DIM, x2, DIM);
}